// papm_convective_flow_71133248357064
// MI455X (gfx1250) — compile-verified
//
#include <hip/hip_runtime.h>
#include <stdint.h>

// Convective-flow stencil, MI455X (gfx1250, wave32).
// Memory-bound: async global->LDS staging (ASYNCcnt path) + f32 VALU FMAs,
// non-temporal output stores to preserve L2 for input halo reuse.

#define IMG     1024
#define IMGMASK 1023
#define TW      64              // tile width  (outputs per block, x)
#define TH      32              // tile height (outputs per block, y)
#define HALO    2
#define LW      (TW + 2 * HALO) // 68 floats = 272 B (16B aligned rows)
#define LH      (TH + 2 * HALO) // 36
#define NTHREADS 256
#define ROWSTEP  (NTHREADS / TW) // 4 row-groups of 64 threads

__device__ __forceinline__ void async_load_f32_to_lds(const float* lds_ptr, const float* gptr) {
    // GLOBAL_LOAD_ASYNC_TO_LDS_B32: VDST = per-lane LDS byte offset,
    // VADDR = 64-bit global address (VGPR pair), SADDR = off. Tracked by ASYNCcnt.
    const uint32_t lds_off = (uint32_t)(uintptr_t)lds_ptr; // low 32 bits of flat shared ptr == LDS offset
    asm volatile("global_load_async_to_lds_b32 %0, %1, off"
                 :: "v"(lds_off), "v"(gptr)
                 : "memory");
}

__device__ __forceinline__ void wait_asynccnt0() {
#if __has_builtin(__builtin_amdgcn_s_wait_asynccnt)
    __builtin_amdgcn_s_wait_asynccnt(0);
#else
    asm volatile("s_wait_asynccnt 0" ::: "memory");
#endif
}

__global__ __launch_bounds__(NTHREADS)
void papm_convective_flow_kernel(const float* __restrict__ x,
                                 const float* __restrict__ wx,
                                 const float* __restrict__ wy,
                                 float* __restrict__ out)
{
    __shared__ float su[LH * LW];
    __shared__ float sv[LH * LW];

    const int tid = threadIdx.x;
    const int tx  = tid & (TW - 1);   // 0..63
    const int ty  = tid >> 6;         // 0..3
    const int x0  = blockIdx.x * TW;
    const int y0  = blockIdx.y * TH;
    const int n   = blockIdx.z;

    const size_t plane = (size_t)IMG * IMG;
    const float* u_base = x + ((size_t)n * 2 + 0) * plane;
    const float* v_base = x + ((size_t)n * 2 + 1) * plane;

    // ---- Stage u and v tiles (periodic-wrap halo) into LDS via async DMA ----
    const int gx_main = (x0 + tx - HALO) & IMGMASK;           // columns 0..63 of LDS row
    const int gx_tail = (x0 + tx + TW - HALO) & IMGMASK;      // columns 64..67 (tx < 4)
    for (int r = ty; r < LH; r += ROWSTEP) {                  // 9 iterations
        const int gy = (y0 + r - HALO) & IMGMASK;
        const float* urow = u_base + (size_t)gy * IMG;
        const float* vrow = v_base + (size_t)gy * IMG;
        float* su_row = &su[r * LW];
        float* sv_row = &sv[r * LW];
        async_load_f32_to_lds(su_row + tx, urow + gx_main);
        async_load_f32_to_lds(sv_row + tx, vrow + gx_main);
        if (tx < LW - TW) {                                   // 4 lanes fill the row tail
            async_load_f32_to_lds(su_row + tx + TW, urow + gx_tail);
            async_load_f32_to_lds(sv_row + tx + TW, vrow + gx_tail);
        }
    }
    wait_asynccnt0();   // this wave's async LDS writes complete
    __syncthreads();    // all waves' tiles visible

    // ---- Stencil coefficients (uniform -> scalar loads) ----
    // wx: (1,1,5,5), nonzero row kh=2 -> flat 10..14 ; c[2] == 0, skipped.
    const float cx0 = wx[10], cx1 = wx[11], cx3 = wx[13], cx4 = wx[14];
    // wy: nonzero column kw=2 -> flat {2,7,17,22}.
    const float cy0 = wy[2],  cy1 = wy[7],  cy3 = wy[17], cy4 = wy[22];

    float* outu = out + ((size_t)n * 2 + 0) * plane;
    float* outv = out + ((size_t)n * 2 + 1) * plane;

    // ---- Compute: 8 rows per thread, fixed column tx ----
    for (int r = ty; r < TH; r += ROWSTEP) {
        const int l0 = (r + HALO) * LW + (tx + HALO);

        const float uc = su[l0];
        const float vc = sv[l0];

        const float u_gx = cx0 * su[l0 - 2]      + cx1 * su[l0 - 1]
                         + cx3 * su[l0 + 1]      + cx4 * su[l0 + 2];
        const float v_gx = cx0 * sv[l0 - 2]      + cx1 * sv[l0 - 1]
                         + cx3 * sv[l0 + 1]      + cx4 * sv[l0 + 2];
        const float u_gy = cy0 * su[l0 - 2 * LW] + cy1 * su[l0 - LW]
                         + cy3 * su[l0 + LW]     + cy4 * su[l0 + 2 * LW];
        const float v_gy = cy0 * sv[l0 - 2 * LW] + cy1 * sv[l0 - LW]
                         + cy3 * sv[l0 + LW]     + cy4 * sv[l0 + 2 * LW];

        const size_t o = (size_t)(y0 + r) * IMG + (x0 + tx);
        __builtin_nontemporal_store(uc * u_gx + vc * u_gy, &outu[o]);
        __builtin_nontemporal_store(uc * v_gx + vc * v_gy, &outv[o]);
    }
}

extern "C" void kernel_launch(void* const* d_in, const int* in_sizes, int n_in,
                              void* d_out, int out_size, void* d_ws, size_t ws_size,
                              hipStream_t stream) {
    (void)in_sizes; (void)n_in; (void)out_size; (void)d_ws; (void)ws_size;
    const float* x  = (const float*)d_in[0];
    const float* wx = (const float*)d_in[1];
    const float* wy = (const float*)d_in[2];
    float* out = (float*)d_out;

    dim3 grid(IMG / TW, IMG / TH, 32);   // (16, 32, 32)
    dim3 block(NTHREADS);
    papm_convective_flow_kernel<<<grid, block, 0, stream>>>(x, wx, wy, out);
}